// Attention_57243324121291
// MI455X (gfx1250) — compile-verified
//
#include <hip/hip_runtime.h>
#include <hip/hip_bf16.h>

// Problem constants (from reference)
#define L_SEQ   2048
#define BATCH   256
#define ENC_H   512
#define DEC_H   512
#define ATTN_H  256

typedef __attribute__((ext_vector_type(2))) float v2f;
typedef __attribute__((ext_vector_type(8))) float v8f;

// ---------------------------------------------------------------------------
// Kernel 1: dec_q[b,a] = sum_h Hdec[b,h] * Wdec[a,h] + bdec[a]
// GEMM M=256(b), N=256(a), K=512(h). One wave per 16x16 tile, WMMA f32 16x16x4.
// A frag (16x4 f32, 2 VGPRs): lane -> M=lane&15, K-pair = (lane>>4)*2  (ISA 7.12.2)
// B frag mirrored: lane -> N=lane&15, K-pair = (lane>>4)*2
// Both A[m,k]=Hdec[m*512+k] and B[k,n]=Wdec[n*512+k] are contiguous in k -> float2 loads.
// ---------------------------------------------------------------------------
__global__ __launch_bounds__(32) void decq_wmma(const float* __restrict__ Hdec,
                                                const float* __restrict__ Wdec,
                                                const float* __restrict__ bdec,
                                                float* __restrict__ decq) {
    const int lane = threadIdx.x;
    const int m0 = blockIdx.y * 16;
    const int n0 = blockIdx.x * 16;
    const int r  = lane & 15;          // M for A-frag, N for B-frag
    const int kb = (lane >> 4) * 2;    // K-pair base within 4-wide K slice

    const float* arow = Hdec + (size_t)(m0 + r) * DEC_H;
    const float* brow = Wdec + (size_t)(n0 + r) * DEC_H;

    v8f acc = {};
    #pragma unroll 4
    for (int k0 = 0; k0 < DEC_H; k0 += 4) {
        v2f a = *(const v2f*)(arow + k0 + kb);
        v2f b = *(const v2f*)(brow + k0 + kb);
        acc = __builtin_amdgcn_wmma_f32_16x16x4_f32(false, a, false, b,
                                                    (short)0, acc, false, false);
    }
    // C/D layout: lane&15 = N; rows M = (lane>>4)*8 + i in VGPR i
    const int N  = lane & 15;
    const int mh = (lane >> 4) * 8;
    const float bias = bdec[n0 + N];
    #pragma unroll
    for (int i = 0; i < 8; ++i)
        decq[(size_t)(m0 + mh + i) * ATTN_H + n0 + N] = acc[i] + bias;
}

// ---------------------------------------------------------------------------
// Kernel 2: v[b,e] = sum_a decq[b,a] * Wenc[a,e]
// GEMM M=256(b), N=512(e), K=256(a). A contiguous in k; B strided (stride 512).
// ---------------------------------------------------------------------------
__global__ __launch_bounds__(32) void v_wmma(const float* __restrict__ decq,
                                             const float* __restrict__ Wenc,
                                             float* __restrict__ v) {
    const int lane = threadIdx.x;
    const int m0 = blockIdx.y * 16;
    const int n0 = blockIdx.x * 16;
    const int r  = lane & 15;
    const int kb = (lane >> 4) * 2;

    const float* arow = decq + (size_t)(m0 + r) * ATTN_H;
    const float* bcol = Wenc + (n0 + r);      // B[k,n] = Wenc[k*512 + n]

    v8f acc = {};
    #pragma unroll 4
    for (int k0 = 0; k0 < ATTN_H; k0 += 4) {
        v2f a = *(const v2f*)(arow + k0 + kb);
        v2f b;
        b.x = bcol[(size_t)(k0 + kb)     * ENC_H];
        b.y = bcol[(size_t)(k0 + kb + 1) * ENC_H];
        acc = __builtin_amdgcn_wmma_f32_16x16x4_f32(false, a, false, b,
                                                    (short)0, acc, false, false);
    }
    const int N  = lane & 15;
    const int mh = (lane >> 4) * 8;
    #pragma unroll
    for (int i = 0; i < 8; ++i)
        v[(size_t)(m0 + mh + i) * ENC_H + n0 + N] = acc[i];
}

// ---------------------------------------------------------------------------
// Kernel 3: c[b] = dot(decq[b,:], benc) -- 64K MACs total, trivial.
// ---------------------------------------------------------------------------
__global__ __launch_bounds__(256) void c_kernel(const float* __restrict__ decq,
                                                const float* __restrict__ benc,
                                                float* __restrict__ c) {
    const int b = threadIdx.x;
    float s = 0.f;
    for (int a = 0; a < ATTN_H; ++a)
        s += decq[(size_t)b * ATTN_H + a] * benc[a];
    c[b] = s;
}

// ---------------------------------------------------------------------------
// Kernel 4 (bandwidth-bound main pass): one block per batch b.
// energy[l] = tanh(enc[l,b,:]·v[b] + c[b]); softmax over l with mask, fused.
// 32 waves/block, each wave: 64 rows, lane-strided float4 loads (2KB/row),
// wave32 butterfly reduction. Softmax via LDS block reduction.
// ---------------------------------------------------------------------------
__global__ __launch_bounds__(1024) void attn_main(const float* __restrict__ enc,
                                                  const float* __restrict__ mask,
                                                  const float* __restrict__ v,
                                                  const float* __restrict__ c,
                                                  float* __restrict__ out) {
    __shared__ float energy[L_SEQ];
    __shared__ float red[32];

    const int b    = blockIdx.x;
    const int tid  = threadIdx.x;
    const int lane = tid & 31;
    const int wave = tid >> 5;

    // v[b] resident in registers: 4 x float4 per lane (512 floats / wave)
    const float4* vb = (const float4*)(v + (size_t)b * ENC_H);
    float4 vf[4];
    #pragma unroll
    for (int i = 0; i < 4; ++i) vf[i] = vb[i * 32 + lane];
    const float cb = c[b];

    // Stream encoder rows: wave w handles l = w, w+32, ...
    for (int l = wave; l < L_SEQ; l += 32) {
        const float4* row = (const float4*)(enc + ((size_t)l * BATCH + b) * ENC_H);
        float acc = 0.f;
        #pragma unroll
        for (int i = 0; i < 4; ++i) {
            float4 e = row[i * 32 + lane];
            acc += e.x * vf[i].x + e.y * vf[i].y + e.z * vf[i].z + e.w * vf[i].w;
        }
        #pragma unroll
        for (int off = 16; off > 0; off >>= 1)
            acc += __shfl_xor(acc, off, 32);
        if (lane == 0) energy[l] = tanhf(acc + cb);
    }
    __syncthreads();

    // Masked softmax over L_SEQ=2048: each thread owns 2 elements.
    const float* mrow = mask + (size_t)b * L_SEQ;
    float x0 = energy[tid]        + mrow[tid];
    float x1 = energy[tid + 1024] + mrow[tid + 1024];

    // block max
    float m = fmaxf(x0, x1);
    #pragma unroll
    for (int off = 16; off > 0; off >>= 1)
        m = fmaxf(m, __shfl_xor(m, off, 32));
    if (lane == 0) red[wave] = m;
    __syncthreads();
    if (wave == 0) {
        float t = red[lane];
        #pragma unroll
        for (int off = 16; off > 0; off >>= 1)
            t = fmaxf(t, __shfl_xor(t, off, 32));
        red[lane] = t;
    }
    __syncthreads();
    m = red[0];

    // block sum of exp
    float e0 = __expf(x0 - m), e1 = __expf(x1 - m);
    float s = e0 + e1;
    #pragma unroll
    for (int off = 16; off > 0; off >>= 1)
        s += __shfl_xor(s, off, 32);
    __syncthreads();                 // everyone has read red[0] before reuse
    if (lane == 0) red[wave] = s;
    __syncthreads();
    if (wave == 0) {
        float t = red[lane];
        #pragma unroll
        for (int off = 16; off > 0; off >>= 1)
            t += __shfl_xor(t, off, 32);
        red[lane] = t;
    }
    __syncthreads();
    const float inv = 1.0f / red[0];

    out[(size_t)b * L_SEQ + tid]        = e0 * inv;
    out[(size_t)b * L_SEQ + tid + 1024] = e1 * inv;
}

// ---------------------------------------------------------------------------
extern "C" void kernel_launch(void* const* d_in, const int* in_sizes, int n_in,
                              void* d_out, int out_size, void* d_ws, size_t ws_size,
                              hipStream_t stream) {
    const float* enc  = (const float*)d_in[0]; // [2048,256,512]
    const float* hdec = (const float*)d_in[1]; // [256,512]
    const float* mask = (const float*)d_in[2]; // [256,2048]
    const float* Wenc = (const float*)d_in[3]; // [256,512]
    const float* benc = (const float*)d_in[4]; // [256]
    const float* Wdec = (const float*)d_in[5]; // [256,512]
    const float* bdec = (const float*)d_in[6]; // [256]
    float* out = (float*)d_out;                // [256,2048,1]

    // Workspace layout (~771 KB): decq | v | c
    float* decq = (float*)d_ws;                    // 256*256
    float* v    = decq + BATCH * ATTN_H;           // 256*512
    float* c    = v    + BATCH * ENC_H;            // 256

    decq_wmma<<<dim3(ATTN_H / 16, BATCH / 16), 32, 0, stream>>>(hdec, Wdec, bdec, decq);
    v_wmma   <<<dim3(ENC_H  / 16, BATCH / 16), 32, 0, stream>>>(decq, Wenc, v);
    c_kernel <<<1, BATCH, 0, stream>>>(decq, benc, c);
    attn_main<<<BATCH, 1024, 0, stream>>>(enc, mask, v, c, out);
}